// GATLayer_5317169512876
// MI455X (gfx1250) — compile-verified
//
#include <hip/hip_runtime.h>
#include <math.h>

// GAT layer for MI455X (gfx1250, wave32).
//   N = 50000 nodes, E = 640000 edges, H = 128 hidden.
// Bandwidth-bound (~0.7-1 GB traffic -> ~40us at 23.3 TB/s HBM).
// GEMM prologue uses V_WMMA_F32_16X16X4_F32 (full f32, matches reference math).

#define H 128
#define NEG_SLOPE 0.01f

typedef __attribute__((ext_vector_type(2))) float v2f;
typedef __attribute__((ext_vector_type(8))) float v8f;

// ---------------------------------------------------------------------------
// Kernel 1: messages[N,H] = node_emb[N,H] @ W1[H,H]^T  via f32 WMMA 16x16x4.
// Block = 256 threads = 8 waves; wave w owns output columns [16w, 16w+16).
// Each wave keeps its whole B fragment (W1^T tile, all K=128) in 64 VGPRs and
// grid-strides over 16-row tiles of A, so W1 is loaded once per wave.
//
// Fragment layouts (ISA 7.12.2, wave32):
//   A 16x4 f32 : lane&15 = M row; lane>=16 selects K pair {2,3} vs {0,1};
//                VGPR0 = even K, VGPR1 = odd K  -> one b64 load per step.
//   B 4x16 f32 : lane&15 = N col; lane>=16 selects K rows {2,3} vs {0,1};
//                B[k][n] = W1[n][k] -> consecutive k contiguous in W1 row.
//   C/D 16x16  : VGPR v -> row v + 8*(lane>=16), col = lane&15.
// ---------------------------------------------------------------------------
__global__ __launch_bounds__(256) void gat_gemm_wmma(
    const float* __restrict__ A, const float* __restrict__ W1,
    float* __restrict__ msgs, int nTiles)
{
  const int lane = threadIdx.x & 31;
  const int wave = threadIdx.x >> 5;
  const int col  = lane & 15;
  const int half = lane >> 4;          // 0 or 1
  const int n0   = wave * 16;          // output column tile

  // Register-resident B fragments for all 32 K-steps (K = 4 per step).
  v2f bf[32];
  const float* w1row = W1 + (size_t)(n0 + col) * H;
  #pragma unroll
  for (int ks = 0; ks < 32; ++ks)
    bf[ks] = *(const v2f*)(w1row + ks * 4 + half * 2);

  for (int mt = blockIdx.x; mt < nTiles; mt += gridDim.x) {
    const int m0 = mt * 16;
    const float* arow = A + (size_t)(m0 + col) * H;
    v8f acc = {};
    #pragma unroll
    for (int ks = 0; ks < 32; ++ks) {
      v2f af = *(const v2f*)(arow + ks * 4 + half * 2);
      acc = __builtin_amdgcn_wmma_f32_16x16x4_f32(
          /*neg_a=*/false, af, /*neg_b=*/false, bf[ks],
          /*c_mod=*/(short)0, acc, /*reuse_a=*/false, /*reuse_b=*/false);
    }
    float* orow = msgs + (size_t)(m0 + half * 8) * H + n0 + col;
    #pragma unroll
    for (int v = 0; v < 8; ++v) orow[(size_t)v * H] = acc[v];
  }
}

// ---------------------------------------------------------------------------
// Kernel 2: per-edge logits + leaky ReLU.  One wave per edge (grid-stride).
// Lane l owns channels [4l, 4l+4): three b128 loads, fused dot, wave32
// xor-shuffle reduction. W2 slices live in registers for the whole loop.
// ---------------------------------------------------------------------------
__global__ __launch_bounds__(256) void gat_edge_logits(
    const float* __restrict__ msgs, const float* __restrict__ eemb,
    const int* __restrict__ sn, const int* __restrict__ en,
    const float* __restrict__ W2, float* __restrict__ logits, int E)
{
  const int lane   = threadIdx.x & 31;
  const int gwave  = blockIdx.x * (blockDim.x >> 5) + (threadIdx.x >> 5);
  const int nWaves = gridDim.x * (blockDim.x >> 5);

  const float4 w2a = *(const float4*)(W2 +         lane * 4);
  const float4 w2b = *(const float4*)(W2 +     H + lane * 4);
  const float4 w2c = *(const float4*)(W2 + 2 * H + lane * 4);

  for (int e = gwave; e < E; e += nWaves) {
    const int si = sn[e];
    const int ei = en[e];
    const float4 mi = *(const float4*)(msgs + (size_t)si * H + lane * 4);
    const float4 mj = *(const float4*)(msgs + (size_t)ei * H + lane * 4);
    const float4 ee = *(const float4*)(eemb + (size_t)e  * H + lane * 4);

    float p = mi.x * w2a.x + mi.y * w2a.y + mi.z * w2a.z + mi.w * w2a.w
            + mj.x * w2b.x + mj.y * w2b.y + mj.z * w2b.z + mj.w * w2b.w
            + ee.x * w2c.x + ee.y * w2c.y + ee.z * w2c.z + ee.w * w2c.w;

    #pragma unroll
    for (int m = 16; m >= 1; m >>= 1) p += __shfl_xor(p, m, 32);

    if (lane == 0)
      logits[e] = (p >= 0.0f) ? p : NEG_SLOPE * p;
  }
}

// ---------------------------------------------------------------------------
// Kernel 3: segment softmax over edges of each start node + weighted scatter
// sum of destination messages. start_nodes is sorted -> binary search gives
// each node's edge range [lo,hi) with no atomics. One wave per node; lane l
// owns output channels [4l, 4l+4).
// ---------------------------------------------------------------------------
__global__ __launch_bounds__(256) void gat_segment_softmax_agg(
    const float* __restrict__ msgs, const int* __restrict__ sn,
    const int* __restrict__ en, const float* __restrict__ logits,
    float* __restrict__ out, int N, int E)
{
  const int lane = threadIdx.x & 31;
  const int node = blockIdx.x * (blockDim.x >> 5) + (threadIdx.x >> 5);
  if (node >= N) return;

  // lower_bound(node) and lower_bound(node+1) over sorted start_nodes.
  int lo;
  {
    int l = 0, r = E;
    while (l < r) { int m = (l + r) >> 1; if (sn[m] < node) l = m + 1; else r = m; }
    lo = l;
  }
  int hi;
  {
    int l = lo, r = E;
    while (l < r) { int m = (l + r) >> 1; if (sn[m] <= node) l = m + 1; else r = m; }
    hi = l;
  }

  float4 acc = make_float4(0.f, 0.f, 0.f, 0.f);
  if (lo == hi) {                      // empty segment -> zeros (sum identity)
    *(float4*)(out + (size_t)node * H + lane * 4) = acc;
    return;
  }

  // Pass 1: segment max (lane-strided + wave32 reduce).
  float mx = -__builtin_inff();
  for (int e = lo + lane; e < hi; e += 32) mx = fmaxf(mx, logits[e]);
  #pragma unroll
  for (int m = 16; m >= 1; m >>= 1) mx = fmaxf(mx, __shfl_xor(mx, m, 32));

  // Pass 2: denominator.
  float den = 0.0f;
  for (int e = lo + lane; e < hi; e += 32) den += __expf(logits[e] - mx);
  #pragma unroll
  for (int m = 16; m >= 1; m >>= 1) den += __shfl_xor(den, m, 32);

  // Pass 3: weighted accumulation of destination messages (coalesced 512B
  // gather per edge; per-lane exp is redundant across lanes but cheap).
  for (int e = lo; e < hi; ++e) {
    const float w = __expf(logits[e] - mx);
    const int  j  = en[e];
    const float4 mj = *(const float4*)(msgs + (size_t)j * H + lane * 4);
    acc.x += w * mj.x; acc.y += w * mj.y; acc.z += w * mj.z; acc.w += w * mj.w;
  }

  const float inv = 1.0f / den;
  float4 r = make_float4(acc.x * inv, acc.y * inv, acc.z * inv, acc.w * inv);
  *(float4*)(out + (size_t)node * H + lane * 4) = r;
}

// ---------------------------------------------------------------------------
// Host launcher. Inputs (setup_inputs order):
//   0: node_embeddings f32 [N*H]   1: edge_embeddings f32 [E*H]
//   2: start_nodes i32 [E]         3: end_nodes i32 [E]
//   4: W1 f32 [H*H]                5: W2 f32 [3H]
// Workspace layout: [ messages N*H f32 | logits E f32 ]  (~28.2 MB).
// ---------------------------------------------------------------------------
extern "C" void kernel_launch(void* const* d_in, const int* in_sizes, int n_in,
                              void* d_out, int out_size, void* d_ws, size_t ws_size,
                              hipStream_t stream) {
  const float* node_emb = (const float*)d_in[0];
  const float* edge_emb = (const float*)d_in[1];
  const int*   sn       = (const int*)d_in[2];
  const int*   en       = (const int*)d_in[3];
  const float* W1       = (const float*)d_in[4];
  const float* W2       = (const float*)d_in[5];
  float* out = (float*)d_out;

  const int N = in_sizes[0] / H;
  const int E = in_sizes[2];

  float* msgs   = (float*)d_ws;
  float* logits = msgs + (size_t)N * H;

  // K1: WMMA GEMM (N assumed multiple of 16; N = 50000 = 3125 * 16).
  const int nTiles = N / 16;
  const int g1 = nTiles < 1024 ? nTiles : 1024;
  gat_gemm_wmma<<<g1, 256, 0, stream>>>(node_emb, W1, msgs, nTiles);

  // K2: edge logits (one wave per edge, grid-stride).
  gat_edge_logits<<<2048, 256, 0, stream>>>(msgs, edge_emb, sn, en, W2, logits, E);

  // K3: segment softmax + weighted scatter (one wave per node).
  const int g3 = (N + 7) / 8;     // 8 waves per 256-thread block
  gat_segment_softmax_agg<<<g3, 256, 0, stream>>>(msgs, sn, en, logits, out, N, E);
}